// E62SelectiveWrite_4887672783140
// MI455X (gfx1250) — compile-verified
//
#include <hip/hip_runtime.h>

// ---------------------------------------------------------------------------
// Problem constants (from reference: B=4, T=4096, D=1024)
// ---------------------------------------------------------------------------
#define B_DIM 4
#define T_LEN 4096
#define D_DIM 1024
#define BD (B_DIM * D_DIM)          // 4096
#define NCHUNK 64                   // scan chunks
#define CLEN (T_LEN / NCHUNK)       // 64 steps per chunk

// GEMM tiling
#define TM 128
#define TN 64
#define BK 32
#define LDS_STRIDE (BK + 4)         // 36 floats -> 144B row stride (16B aligned)

typedef float v2f __attribute__((ext_vector_type(2)));
typedef float v8f __attribute__((ext_vector_type(8)));

enum Epilogue { EP_NONE = 0, EP_SILU = 1, EP_SIGMOID_BIAS = 2, EP_TANH_BIAS = 3 };

// ---- gfx1250 async global->LDS path (guarded; fallback = batched sync) ----
#if defined(__has_builtin)
#if __has_builtin(__builtin_amdgcn_global_load_async_to_lds_b128) && \
    __has_builtin(__builtin_amdgcn_s_wait_asynccnt)
#define USE_ASYNC_LDS 1
#endif
#endif
#ifndef USE_ASYNC_LDS
#define USE_ASYNC_LDS 0
#endif

#if USE_ASYNC_LDS
// Builtin signature (from hipcc diagnostic): param 0 is
//   '__attribute__((__vector_size__(4*sizeof(int)))) int __device__ *'  (AS1 int4*)
// so use gcc-style generic vector pointees in the right address spaces.
typedef int gvec4i __attribute__((vector_size(16)));
typedef __attribute__((address_space(1))) gvec4i glb_v4i_t;
typedef __attribute__((address_space(3))) gvec4i lds_v4i_t;

__device__ __forceinline__ void async_cp16(const float* g, float* l) {
    __builtin_amdgcn_global_load_async_to_lds_b128(
        (glb_v4i_t*)(unsigned long long)g,
        (lds_v4i_t*)(unsigned int)(unsigned long long)l,
        /*offset=*/0, /*cpol=*/0);
}
#endif

__device__ __forceinline__ float sigmoid_f(float x) {
    return 1.0f / (1.0f + __expf(-x));
}
__device__ __forceinline__ float tanh_f(float x) {
    float e = __expf(-2.0f * x);
    return (1.0f - e) / (1.0f + e);
}

// ---------------------------------------------------------------------------
// Tiled fp32 GEMM on the WMMA pipe:  C[M,N] = act(A[M,K] * W[N,K]^T + bias)
// 256 threads (8 wave32), block tile 128(M) x 64(N), K staged 32 at a time
// into double-buffered LDS via async global->LDS copies. Each wave computes
// a 32x32 C tile = 2x2 grid of 16x16 V_WMMA_F32_16X16X4_F32 accumulators.
// ---------------------------------------------------------------------------
template <int EPI>
__global__ __launch_bounds__(256) void gemm_wmma_f32(
    const float* __restrict__ A,     // [M, K] row-major
    const float* __restrict__ W,     // [N, K] row-major
    const float* __restrict__ bias,  // [N] or nullptr
    float* __restrict__ C,           // [M, N]
    int M, int N, int K)
{
    __shared__ float lds_a[2][TM * LDS_STRIDE];
    __shared__ float lds_b[2][TN * LDS_STRIDE];

    const int tid  = threadIdx.x;
    const int lane = tid & 31;
    const int wave = tid >> 5;       // 0..7
    const int wm   = wave & 3;       // M quadrant of block tile (32 rows each)
    const int wn   = wave >> 2;      // N half of block tile (32 cols each)
    const int rl   = lane & 15;      // row/col within 16
    const int kh   = lane >> 4;      // K-half selector (lanes 16..31 hold K+2)

    const int m0 = blockIdx.y * TM;
    const int n0 = blockIdx.x * TN;

    const int r_ld  = tid >> 3;          // 0..31: row within a 32-row slab
    const int c4_ld = (tid & 7) << 2;    // 0..28: float4 column offset

    v8f acc[2][2];
    #pragma unroll
    for (int i = 0; i < 2; ++i)
        #pragma unroll
        for (int j = 0; j < 2; ++j)
            acc[i][j] = (v8f){0.f, 0.f, 0.f, 0.f, 0.f, 0.f, 0.f, 0.f};

    // ---- stage one K-panel (A: 128x32, W: 64x32) into LDS buffer `buf` ----
    auto stage = [&](int buf, int k0) {
#if USE_ASYNC_LDS
        #pragma unroll
        for (int i = 0; i < 4; ++i) {
            const int row = r_ld + i * 32;
            async_cp16(A + (size_t)(m0 + row) * K + k0 + c4_ld,
                       &lds_a[buf][row * LDS_STRIDE + c4_ld]);
        }
        #pragma unroll
        for (int i = 0; i < 2; ++i) {
            const int row = r_ld + i * 32;
            async_cp16(W + (size_t)(n0 + row) * K + k0 + c4_ld,
                       &lds_b[buf][row * LDS_STRIDE + c4_ld]);
        }
#else
        float4 ra[4], rb[2];
        #pragma unroll
        for (int i = 0; i < 4; ++i)
            ra[i] = *(const float4*)(A + (size_t)(m0 + r_ld + i * 32) * K + k0 + c4_ld);
        #pragma unroll
        for (int i = 0; i < 2; ++i)
            rb[i] = *(const float4*)(W + (size_t)(n0 + r_ld + i * 32) * K + k0 + c4_ld);
        #pragma unroll
        for (int i = 0; i < 4; ++i)
            *(float4*)(&lds_a[buf][(r_ld + i * 32) * LDS_STRIDE + c4_ld]) = ra[i];
        #pragma unroll
        for (int i = 0; i < 2; ++i)
            *(float4*)(&lds_b[buf][(r_ld + i * 32) * LDS_STRIDE + c4_ld]) = rb[i];
#endif
    };

    stage(0, 0);  // prologue fill of buffer 0

    for (int k0 = 0; k0 < K; k0 += BK) {
        const int cur = (k0 / BK) & 1;
#if USE_ASYNC_LDS
        __builtin_amdgcn_s_wait_asynccnt(0);   // own async copies into `cur` done
#endif
        __syncthreads();                       // all waves' copies visible; all
                                               // waves done reading `cur^1`
        if (k0 + BK < K) stage(cur ^ 1, k0 + BK);  // overlap next panel with WMMAs

        // ---- 8 x (2x2) WMMA steps over this K panel ----
        #pragma unroll
        for (int kk = 0; kk < BK; kk += 4) {
            v2f afrag[2], bfrag[2];
            #pragma unroll
            for (int ms = 0; ms < 2; ++ms)
                afrag[ms] = *(const v2f*)(&lds_a[cur][(wm * 32 + ms * 16 + rl) * LDS_STRIDE + kk + kh * 2]);
            #pragma unroll
            for (int ns = 0; ns < 2; ++ns)
                bfrag[ns] = *(const v2f*)(&lds_b[cur][(wn * 32 + ns * 16 + rl) * LDS_STRIDE + kk + kh * 2]);

            #pragma unroll
            for (int ms = 0; ms < 2; ++ms)
                #pragma unroll
                for (int ns = 0; ns < 2; ++ns)
                    acc[ms][ns] = __builtin_amdgcn_wmma_f32_16x16x4_f32(
                        /*neg_a=*/false, afrag[ms],
                        /*neg_b=*/false, bfrag[ns],
                        /*c_mod=*/(short)0, acc[ms][ns],
                        /*reuse_a=*/false, /*reuse_b=*/false);
        }
    }

    // ---- fused epilogue + store (C layout: N=lane&15, M = (lane>>4)*8 + vgpr) ----
    #pragma unroll
    for (int ms = 0; ms < 2; ++ms) {
        #pragma unroll
        for (int ns = 0; ns < 2; ++ns) {
            const int col = n0 + wn * 32 + ns * 16 + rl;
            float bcol = 0.0f;
            if (EPI == EP_SIGMOID_BIAS || EPI == EP_TANH_BIAS) bcol = bias[col];
            #pragma unroll
            for (int i = 0; i < 8; ++i) {
                const int row = m0 + wm * 32 + ms * 16 + kh * 8 + i;
                float x = acc[ms][ns][i];
                if (EPI == EP_SILU)              x = x * sigmoid_f(x);
                else if (EPI == EP_SIGMOID_BIAS) x = sigmoid_f(x + bcol);
                else if (EPI == EP_TANH_BIAS)    x = tanh_f(x + bcol);
                C[(size_t)row * N + col] = x;
            }
        }
    }
}

// ---------------------------------------------------------------------------
// Chunked parallel scan of  h_t = (1-k_t) h_{t-1} + k_t v_t   (linear in h)
// k, v, out laid out [B, T, D] (same layout the GEMMs produce/consume).
// ---------------------------------------------------------------------------

// Pass 1: per (chunk, b, d) compute (prod a, local scan end) with h_start = 0.
__global__ __launch_bounds__(256) void scan_pass1(
    const float* __restrict__ Kb, const float* __restrict__ Vb,
    float* __restrict__ Aend, float* __restrict__ Bend)
{
    const int idx = blockIdx.x * blockDim.x + threadIdx.x; // [NCHUNK * BD]
    const int r = idx & (BD - 1);
    const int c = idx >> 12;            // / BD
    const int b = r >> 10;              // / D_DIM
    const int d = r & (D_DIM - 1);

    const size_t base = (size_t)b * T_LEN * D_DIM + (size_t)(c * CLEN) * D_DIM + d;
    const float* kp = Kb + base;
    const float* vp = Vb + base;

    float aprod = 1.0f, bacc = 0.0f;
    #pragma unroll 4
    for (int t = 0; t < CLEN; ++t) {
        const float kk = kp[(size_t)t * D_DIM];
        const float vv = vp[(size_t)t * D_DIM];
        const float a = 1.0f - kk;
        bacc = a * bacc + kk * vv;
        aprod *= a;
    }
    Aend[idx] = aprod;
    Bend[idx] = bacc;
}

// Pass 2: tiny sequential pass over the 64 chunk summaries (L2-resident).
__global__ __launch_bounds__(256) void scan_pass2(
    const float* __restrict__ h0,
    const float* __restrict__ Aend, const float* __restrict__ Bend,
    float* __restrict__ Hstart, float* __restrict__ hfinal)
{
    const int r = blockIdx.x * blockDim.x + threadIdx.x; // [BD]
    float h = h0[r];
    #pragma unroll 4
    for (int c = 0; c < NCHUNK; ++c) {
        Hstart[c * BD + r] = h;
        h = Aend[c * BD + r] * h + Bend[c * BD + r];
    }
    hfinal[r] = h;
}

// Pass 3: replay each chunk with correct h_start; apply gating h^2*sigmoid(h).
__global__ __launch_bounds__(256) void scan_pass3(
    const float* __restrict__ Kb, const float* __restrict__ Vb,
    const float* __restrict__ Hstart, float* __restrict__ Out)
{
    const int idx = blockIdx.x * blockDim.x + threadIdx.x; // [NCHUNK * BD]
    const int r = idx & (BD - 1);
    const int c = idx >> 12;
    const int b = r >> 10;
    const int d = r & (D_DIM - 1);

    const size_t base = (size_t)b * T_LEN * D_DIM + (size_t)(c * CLEN) * D_DIM + d;
    const float* kp = Kb + base;
    const float* vp = Vb + base;
    float* op = Out + base;

    float h = Hstart[idx];
    #pragma unroll 4
    for (int t = 0; t < CLEN; ++t) {
        const float kk = kp[(size_t)t * D_DIM];
        const float vv = vp[(size_t)t * D_DIM];
        h = (1.0f - kk) * h + kk * vv;
        op[(size_t)t * D_DIM] = h * h * sigmoid_f(h);
    }
}

// ---------------------------------------------------------------------------
// Launch
// ---------------------------------------------------------------------------
extern "C" void kernel_launch(void* const* d_in, const int* in_sizes, int n_in,
                              void* d_out, int out_size, void* d_ws, size_t ws_size,
                              hipStream_t stream) {
    (void)in_sizes; (void)n_in; (void)out_size; (void)ws_size;

    const float* x     = (const float*)d_in[0];  // [B,T,D]
    const float* h0    = (const float*)d_in[1];  // [B,D]
    const float* W_in  = (const float*)d_in[2];  // [D,D]
    const float* W_k   = (const float*)d_in[3];
    const float* b_k   = (const float*)d_in[4];
    const float* W_v   = (const float*)d_in[5];
    const float* b_v   = (const float*)d_in[6];
    const float* W_out = (const float*)d_in[7];

    float* y      = (float*)d_out;                       // [B,T,D]
    float* hfinal = y + (size_t)B_DIM * T_LEN * D_DIM;   // [B,D]

    const size_t BTD = (size_t)B_DIM * T_LEN * D_DIM;
    float* ws    = (float*)d_ws;
    float* xp    = ws;               // [B,T,D] silu(x@W_in^T); reused for gated out
    float* kbuf  = ws + BTD;         // [B,T,D]
    float* vbuf  = ws + 2 * BTD;     // [B,T,D]
    float* Aend  = ws + 3 * BTD;     // [NCHUNK, BD]
    float* Bend  = Aend + (size_t)NCHUNK * BD;
    float* Hst   = Bend + (size_t)NCHUNK * BD;

    const int M = B_DIM * T_LEN;     // 16384
    const int N = D_DIM, K = D_DIM;  // 1024

    dim3 blk(256);
    dim3 gg(N / TN, M / TM);         // 16 x 128 blocks

    // 1) xp = silu(x @ W_in^T)
    gemm_wmma_f32<EP_SILU><<<gg, blk, 0, stream>>>(x, W_in, nullptr, xp, M, N, K);
    // 2) k = sigmoid(xp @ W_k^T + b_k)
    gemm_wmma_f32<EP_SIGMOID_BIAS><<<gg, blk, 0, stream>>>(xp, W_k, b_k, kbuf, M, N, K);
    // 3) v = tanh(xp @ W_v^T + b_v)
    gemm_wmma_f32<EP_TANH_BIAS><<<gg, blk, 0, stream>>>(xp, W_v, b_v, vbuf, M, N, K);

    // 4) chunked linear scan + gating (xp buffer is dead -> reuse as gated out)
    scan_pass1<<<(NCHUNK * BD) / 256, blk, 0, stream>>>(kbuf, vbuf, Aend, Bend);
    scan_pass2<<<BD / 256, blk, 0, stream>>>(h0, Aend, Bend, Hst, hfinal);
    scan_pass3<<<(NCHUNK * BD) / 256, blk, 0, stream>>>(kbuf, vbuf, Hst, xp);

    // 5) y = out @ W_out^T
    gemm_wmma_f32<EP_NONE><<<gg, blk, 0, stream>>>(xp, W_out, nullptr, y, M, N, K);
}